// CalcImpute_25443386261851
// MI455X (gfx1250) — compile-verified
//
#include <hip/hip_runtime.h>
#include <stdint.h>

// ---------------- problem constants (from reference) ----------------
#define N_DONORS      100000
#define KSEL          5
#define BLOCK         256
#define WAVE_SZ       32
#define NWAVES        (BLOCK / WAVE_SZ)            // 8
#define CHUNK         (N_DONORS / NWAVES)          // 12500 per wave
#define ISSUES        4                            // b128 async issues per tile per wave
#define TILE_FLOATS   (ISSUES * WAVE_SZ * 4)       // 512 floats per tile
#define NBUF          3                            // triple buffering
#define FULL_ITERS    (CHUNK / TILE_FLOATS)        // 24
#define TAIL_START    (FULL_ITERS * TILE_FLOATS)   // 12288 (tail = 212)

// value part = +INF, index part = all-ones (never a real donor)
#define INF_KEY ((((unsigned long long)0x7F800000u) << 32) | 0xFFFFFFFFull)

// ---------------- CDNA5 async global->LDS copy path ----------------
typedef int v4i __attribute__((ext_vector_type(4)));
typedef __attribute__((address_space(1))) v4i gv4i;   // global int4
typedef __attribute__((address_space(3))) v4i lv4i;   // LDS int4

#if defined(__has_builtin) && __has_builtin(__builtin_amdgcn_global_load_async_to_lds_b128)
#define ASYNC_COPY_B128(gsrc, ldst)                                            \
  __builtin_amdgcn_global_load_async_to_lds_b128(                              \
      (gv4i*)(unsigned long long)(gsrc),                                       \
      (lv4i*)(ldst), 0, 0)
#else
#define ASYNC_COPY_B128(gsrc, ldst)                                            \
  asm volatile("global_load_async_to_lds_b128 %0, %1, off"                     \
               :: "v"((unsigned)(unsigned long long)(ldst)),                   \
                  "v"((const void*)(gsrc))                                     \
               : "memory")
#endif

#if defined(__has_builtin) && __has_builtin(__builtin_amdgcn_s_wait_asynccnt)
#define WAIT_ASYNC(n) do { __builtin_amdgcn_s_wait_asynccnt(n);                \
                           asm volatile("" ::: "memory"); } while (0)
#else
#define WAIT_ASYNC(n) asm volatile("s_wait_asynccnt %0" :: "i"(n) : "memory")
#endif

// ---------------- kernel ----------------
__global__ __launch_bounds__(BLOCK)
void knn_impute_col_kernel(const float* __restrict__ dist,   // [rows, N_DONORS]
                           const float* __restrict__ fitx,   // [N_DONORS]
                           const int*   __restrict__ mask,   // [N_DONORS]
                           float* __restrict__ out)          // [rows]
{
    __shared__ __align__(16) float s_tiles[NWAVES][NBUF][TILE_FLOATS]; // 48 KB
    __shared__ unsigned long long  s_red[BLOCK * KSEL];                // 10 KB

    const int row  = blockIdx.x;
    const int tid  = threadIdx.x;
    const int wave = tid >> 5;
    const int lane = tid & 31;

    const float* rowp  = dist + (size_t)row * N_DONORS;
    const float* base  = rowp + wave * CHUNK;    // this wave's private chunk
    const int    gbase = wave * CHUNK;           // donor index of chunk start

    // sorted ascending; key = (float_bits << 32) | donor_index  (dist >= 0 => monotonic)
    unsigned long long top[KSEL];
#pragma unroll
    for (int i = 0; i < KSEL; ++i) top[i] = INF_KEY;
    float thresh = __builtin_inff();             // value of current 5th smallest

    // key-exact insert (rare path); caller guarantees v < thresh was true once,
    // but re-check against the *current* 5th key (earlier inserts in the same
    // group may have lowered it).
    auto ins = [&](float v, int idx) {
        unsigned long long key =
            ((unsigned long long)__float_as_uint(v) << 32) | (unsigned)idx;
        if (key < top[KSEL - 1]) {
            top[KSEL - 1] = key;
#pragma unroll
            for (int i = KSEL - 1; i > 0; --i) {
                unsigned long long a = top[i - 1], b = top[i];
                top[i - 1] = a < b ? a : b;
                top[i]     = a < b ? b : a;
            }
        }
    };

    auto issue_tile = [&](int buf, int off) {
#pragma unroll
        for (int q = 0; q < ISSUES; ++q) {
            const float* g = base + off + q * (WAVE_SZ * 4) + lane * 4;
            float*       l = &s_tiles[wave][buf][q * (WAVE_SZ * 4) + lane * 4];
            ASYNC_COPY_B128(g, l);
        }
    };

    // ---------- triple-buffered async streaming over this wave's chunk ----------
    issue_tile(0, 0);
    issue_tile(1, TILE_FLOATS);
    for (int it = 0; it < FULL_ITERS; ++it) {
        const int buf = it % NBUF;
        if (it + 2 < FULL_ITERS) {
            issue_tile((it + 2) % NBUF, (it + 2) * TILE_FLOATS);
            WAIT_ASYNC(2 * ISSUES);   // two newest tiles may fly; current is done
        } else if (it + 1 < FULL_ITERS) {
            WAIT_ASYNC(ISSUES);
        } else {
            WAIT_ASYNC(0);
        }
#pragma unroll
        for (int q = 0; q < ISSUES; ++q) {
            const float4 v =
                *(const float4*)&s_tiles[wave][buf][q * (WAVE_SZ * 4) + lane * 4];
            const int gi = gbase + it * TILE_FLOATS + q * (WAVE_SZ * 4) + lane * 4;
            // hot path: one float compare per element against cached threshold.
            // stream is in increasing index order, so strict '<' preserves the
            // lowest-index tie-break exactly; NaN fails the compare (weight 0
            // entries can never be among the 5 smallest finite keys).
            const bool c0 = v.x < thresh;
            const bool c1 = v.y < thresh;
            const bool c2 = v.z < thresh;
            const bool c3 = v.w < thresh;
            if (c0 | c1 | c2 | c3) {          // s_cbranch_execz skips when rare
                if (c0) ins(v.x, gi + 0);
                if (c1) ins(v.y, gi + 1);
                if (c2) ins(v.z, gi + 2);
                if (c3) ins(v.w, gi + 3);
                thresh = __uint_as_float((unsigned)(top[KSEL - 1] >> 32));
            }
        }
    }
    // tail (212 elements per wave) straight from global
    for (int j = TAIL_START + lane; j < CHUNK; j += WAVE_SZ) {
        const float v = base[j];
        if (v < thresh) {
            ins(v, gbase + j);
            thresh = __uint_as_float((unsigned)(top[KSEL - 1] >> 32));
        }
    }

    // ---------- block-wide merge of sorted 5-lists (exact k-smallest) ----------
#pragma unroll
    for (int i = 0; i < KSEL; ++i) s_red[tid * KSEL + i] = top[i];
    __syncthreads();

    for (int s = BLOCK / 2; s >= 1; s >>= 1) {
        if (tid < s) {
#pragma unroll
            for (int i = 0; i < KSEL; ++i) {
                unsigned long long key = s_red[(tid + s) * KSEL + i];
                if (key < top[KSEL - 1]) {
                    top[KSEL - 1] = key;
#pragma unroll
                    for (int m = KSEL - 1; m > 0; --m) {
                        unsigned long long a = top[m - 1], b = top[m];
                        top[m - 1] = a < b ? a : b;
                        top[m]     = a < b ? b : a;
                    }
                }
            }
#pragma unroll
            for (int i = 0; i < KSEL; ++i) s_red[tid * KSEL + i] = top[i];
        }
        __syncthreads();
    }

    // ---------- final weighted average (uniform weights, NaN & mask zeroed) ----------
    if (tid == 0) {
        float num = 0.f, den = 0.f;
#pragma unroll
        for (int i = 0; i < KSEL; ++i) {
            const unsigned idx = (unsigned)(top[i] & 0xffffffffull);
            if (idx < (unsigned)N_DONORS) {
                const float d    = __uint_as_float((unsigned)(top[i] >> 32));
                const float wnan = (d != d) ? 0.f : 1.f;          // isnan -> 0
                const float w    = wnan * (float)(1 - mask[idx]); // valid-donor mask
                num += fitx[idx] * w;
                den += w;
            }
        }
        if (den == 0.f) den = 1.f;
        out[row] = num / den;
    }
}

// ---------------- harness entry ----------------
extern "C" void kernel_launch(void* const* d_in, const int* in_sizes, int n_in,
                              void* d_out, int out_size, void* d_ws, size_t ws_size,
                              hipStream_t stream) {
    (void)in_sizes; (void)n_in; (void)d_ws; (void)ws_size;
    const float* dist = (const float*)d_in[0];   // [rows, 100000] f32
    const float* fitx = (const float*)d_in[1];   // [100000] f32
    const int*   mask = (const int*)d_in[2];     // [100000] i32 (JAX x64 off)
    // d_in[3] = n_neighbors scalar; K=5 baked in.
    float* out = (float*)d_out;
    const int rows = out_size;                   // 4096

    hipLaunchKernelGGL(knn_impute_col_kernel, dim3(rows), dim3(BLOCK), 0, stream,
                       dist, fitx, mask, out);
}